// HomographyNet_14078902797002
// MI455X (gfx1250) — compile-verified
//
#include <hip/hip_runtime.h>

// CDNA5 / gfx1250: wave32, WMMA 16x16 tiles.
typedef __attribute__((ext_vector_type(2))) float v2f;   // A/B operand of wmma f32 16x16x4 (2 VGPRs)
typedef __attribute__((ext_vector_type(8))) float v8f;   // C/D accumulator (8 VGPRs)

#define WPB  8     // waves per block (256 threads)
#define SLOT 13    // LDS floats per item slot in phase 1 (gcd(13,64)=1 -> conflict-free gather)

__global__ __launch_bounds__(WPB * 32)
void homog_expm_wmma(const float* __restrict__ V,   // [N,8]  coefficients v
                     const float* __restrict__ Bg,  // [8,9]  sl(3) basis (row-major 3x3 flattened)
                     float* __restrict__ Out,       // [N,9]  H
                     long long Nitems)
{
    __shared__ float lds[WPB * 32 * SLOT];

    const int lane = threadIdx.x & 31;
    const int wib  = threadIdx.x >> 5;
    const long long nWaves = (Nitems + 31) >> 5;
    const long long w = (long long)blockIdx.x * WPB + wib;
    if (w >= nWaves) return;                 // wave-uniform exit; active waves have full EXEC

    float* wls = lds + wib * (32 * SLOT);    // this wave's private LDS region
    const long long base = w << 5;           // first item of this wave (32 items/wave)

    const int  half = lane >> 4;             // 0: lanes 0-15, 1: lanes 16-31
    const int  n    = lane & 15;             // A-row (M) / B-col (N) index for this lane
    const bool nv   = (n < 9);               // only 9 of 16 GEMM columns are real

    // ---------------- B operand: Bmat[8][16] = basis rows, zero-padded 9->16 cols.
    // 32-bit B layout (mirror of A): VGPR0 holds K = 2*half, VGPR1 holds K = 2*half+1.
    v2f bLo, bHi;
    {
        const int k0 = 2 * half;
        bLo.x = nv ? Bg[(k0 + 0) * 9 + n] : 0.0f;
        bLo.y = nv ? Bg[(k0 + 1) * 9 + n] : 0.0f;
        bHi.x = nv ? Bg[(k0 + 4) * 9 + n] : 0.0f;   // K = 4 + 2*half
        bHi.y = nv ? Bg[(k0 + 5) * 9 + n] : 0.0f;   // K = 5 + 2*half
    }

    // ---------------- Phase 1: C = V(16x8) x Bmat(8x16) per 16-item tile, two tiles.
    // K=8 done as two chained K=4 WMMAs. D scattered to LDS one row per item.
#pragma unroll
    for (int t = 0; t < 2; ++t) {
        long long m = base + 16 * t + n;
        if (m > Nitems - 1) m = Nitems - 1;          // tail clamp via select (EXEC untouched)
        const float* vp = V + m * 8;
        v2f aLo = *(const v2f*)(vp + 2 * half);      // v[m][2h], v[m][2h+1]   (K=0..3 slice)
        v2f aHi = *(const v2f*)(vp + 4 + 2 * half);  // v[m][4+2h], v[m][5+2h] (K=4..7 slice)
        v8f acc = {};
        acc = __builtin_amdgcn_wmma_f32_16x16x4_f32(false, aLo, false, bLo,
                                                    (short)0, acc, false, false);
        acc = __builtin_amdgcn_wmma_f32_16x16x4_f32(false, aHi, false, bHi,
                                                    (short)0, acc, false, false);
        if (nv) {
            // D layout: VGPR r = row (r + 8*half) of the tile, column n.
#pragma unroll
            for (int r = 0; r < 8; ++r)
                wls[(16 * t + r + 8 * half) * SLOT + n] = acc[r];
        }
    }

    asm volatile("s_wait_dscnt 0x0" ::: "memory");   // wave-local LDS write->read ordering

    // ---------------- Phase 2: per-lane item, closed-form truncated expm.
    const float e0 = wls[lane * SLOT + 0], e1 = wls[lane * SLOT + 1], e2 = wls[lane * SLOT + 2];
    const float e3 = wls[lane * SLOT + 3], e4 = wls[lane * SLOT + 4], e5 = wls[lane * SLOT + 5];
    const float e6 = wls[lane * SLOT + 6], e7 = wls[lane * SLOT + 7], e8 = wls[lane * SLOT + 8];

    // C is traceless: C^3 = p*C + q*I,  p = 0.5*tr(C^2), q = det(C).
    const float p = 0.5f * (e0 * e0 + e4 * e4 + e8 * e8)
                  + (e1 * e3 + e2 * e6 + e5 * e7);
    const float q = e0 * (e4 * e8 - e5 * e7)
                  - e1 * (e3 * e8 - e5 * e6)
                  + e2 * (e3 * e7 - e4 * e6);

    // Reduce sum_{i=0..9} x^i / i!  mod (x^3 - p x - q):  x^i = al + be*x + ga*x^2
    float ka = 1.0f, kb = 0.0f, kc = 0.0f;          // accumulated (I, C, C^2) coefficients
    float al = 1.0f, be = 0.0f, ga = 0.0f;
    const float wgt[10] = {1.0f, 1.0f, 0.5f, 1.0f/6.0f, 1.0f/24.0f, 1.0f/120.0f,
                           1.0f/720.0f, 1.0f/5040.0f, 1.0f/40320.0f, 1.0f/362880.0f};
#pragma unroll
    for (int i = 1; i <= 9; ++i) {
        const float na = q * ga;
        const float nb = al + p * ga;
        const float ng = be;
        al = na; be = nb; ga = ng;
        ka += wgt[i] * al; kb += wgt[i] * be; kc += wgt[i] * ga;
    }

    // C^2 (full 3x3 square)
    const float d0 = e0 * e0 + e1 * e3 + e2 * e6;
    const float d1 = e0 * e1 + e1 * e4 + e2 * e7;
    const float d2 = e0 * e2 + e1 * e5 + e2 * e8;
    const float d3 = e3 * e0 + e4 * e3 + e5 * e6;
    const float d4 = e3 * e1 + e4 * e4 + e5 * e7;
    const float d5 = e3 * e2 + e4 * e5 + e5 * e8;
    const float d6 = e6 * e0 + e7 * e3 + e8 * e6;
    const float d7 = e6 * e1 + e7 * e4 + e8 * e7;
    const float d8 = e6 * e2 + e7 * e5 + e8 * e8;

    float h[9];
    h[0] = ka + kb * e0 + kc * d0;
    h[1] =      kb * e1 + kc * d1;
    h[2] =      kb * e2 + kc * d2;
    h[3] =      kb * e3 + kc * d3;
    h[4] = ka + kb * e4 + kc * d4;
    h[5] =      kb * e5 + kc * d5;
    h[6] =      kb * e6 + kc * d6;
    h[7] =      kb * e7 + kc * d7;
    h[8] = ka + kb * e8 + kc * d8;

    // ---------------- Phase 3: stage H densely in LDS, then coalesced global stores.
    asm volatile("s_wait_dscnt 0x0" ::: "memory");   // reads above done before overwrite
#pragma unroll
    for (int e = 0; e < 9; ++e)
        wls[lane * 9 + e] = h[e];                    // dense [item*9 + e], stride-9 writes
    asm volatile("s_wait_dscnt 0x0" ::: "memory");

    const long long obase = base * 9;                // wave covers 288 contiguous floats
    const long long total = Nitems * 9;
#pragma unroll
    for (int j = 0; j < 9; ++j) {
        const long long gi = obase + 32 * j + lane;  // fully coalesced b32 stores
        if (gi < total)
            Out[gi] = wls[32 * j + lane];
    }
}

extern "C" void kernel_launch(void* const* d_in, const int* in_sizes, int n_in,
                              void* d_out, int out_size, void* d_ws, size_t ws_size,
                              hipStream_t stream) {
    (void)n_in; (void)out_size; (void)d_ws; (void)ws_size;
    const float* V  = (const float*)d_in[0];   // v: [N,8,1,1]
    const float* Bg = (const float*)d_in[1];   // B: [8,3,3]
    float* Out = (float*)d_out;                // H: [N,3,3]

    const long long N = (long long)in_sizes[0] / 8;
    const long long nWaves = (N + 31) / 32;
    const int blocks = (int)((nWaves + WPB - 1) / WPB);
    homog_expm_wmma<<<blocks, WPB * 32, 0, stream>>>(V, Bg, Out, N);
}